// GCN_87737591923199
// MI455X (gfx1250) — compile-verified
//
#include <hip/hip_runtime.h>

#define HIDDEN   128
#define IN_DIM   9
#define N_GRAPHS 1024
#define BN_EPS   1e-5f

typedef __attribute__((ext_vector_type(2))) float v2f;
typedef __attribute__((ext_vector_type(8))) float v8f;

// ---------------- init: deg=1 (self loop), pooled=0 ----------------
__global__ void init_kernel(float* __restrict__ deg, float* __restrict__ pooled,
                            int n, int pg) {
    int i = blockIdx.x * blockDim.x + threadIdx.x;
    if (i < n)  deg[i] = 1.0f;
    if (i < pg) pooled[i] = 0.0f;
}

// ---------------- degree count over dst ----------------
__global__ void deg_kernel(const long long* __restrict__ dst, float* __restrict__ deg, int e) {
    int i = blockIdx.x * blockDim.x + threadIdx.x;
    if (i < e) atomicAdd(&deg[(int)dst[i]], 1.0f);
}

// ---------------- dis = rsqrt(deg), in place ----------------
__global__ void dis_kernel(float* __restrict__ deg, int n) {
    int i = blockIdx.x * blockDim.x + threadIdx.x;
    if (i < n) deg[i] = rsqrtf(deg[i]);
}

// ---------------- GEMM1: [n,9]x[9,128] -> xw ----------------
__global__ __launch_bounds__(128) void gemm1_kernel(const float* __restrict__ x,
                                                    const float* __restrict__ w1,
                                                    float* __restrict__ xw, int n) {
    __shared__ float lw[IN_DIM * HIDDEN];
    __shared__ float lx[8 * IN_DIM];
    int t = threadIdx.x;
    for (int i = t; i < IN_DIM * HIDDEN; i += 128) lw[i] = w1[i];
    int node0 = blockIdx.x * 8;
    if (t < 8 * IN_DIM) {
        int node = node0 + t / IN_DIM;
        lx[t] = (node < n) ? x[(size_t)node0 * IN_DIM + t] : 0.0f;
    }
    __syncthreads();
    for (int r = 0; r < 8; ++r) {
        int node = node0 + r;
        if (node >= n) break;
        float acc = 0.0f;
#pragma unroll
        for (int k = 0; k < IN_DIM; ++k)
            acc = fmaf(lx[r * IN_DIM + k], lw[k * HIDDEN + t], acc);
        xw[(size_t)node * HIDDEN + t] = acc;
    }
}

// ---------------- scatter init: out = bias + (1/deg)*xw  (self loop + bias) ----------------
__global__ void scat_init_kernel(float* __restrict__ out, const float* __restrict__ xw,
                                 const float* __restrict__ dis, const float* __restrict__ bias,
                                 int n) {
    int i = blockIdx.x * blockDim.x + threadIdx.x;
    if (i >= n * HIDDEN) return;
    int node = i >> 7, h = i & 127;
    float d = dis[node];
    out[i] = fmaf(d * d, xw[i], bias[h]);
}

// ---------------- scatter edges: out[d] += dis[s]*dis[d]*xw[s] ----------------
__global__ void scat_edge_kernel(float* __restrict__ out, const float* __restrict__ xw,
                                 const float* __restrict__ dis,
                                 const long long* __restrict__ src,
                                 const long long* __restrict__ dst, int e) {
    long long idx = (long long)blockIdx.x * blockDim.x + threadIdx.x;
    if (idx >= (long long)e * 32) return;
    int ed = (int)(idx >> 5);
    int c  = (int)(idx & 31);
    int s = (int)src[ed], d = (int)dst[ed];
    float coef = dis[s] * dis[d];
    float4 v = ((const float4*)(xw + (size_t)s * HIDDEN))[c];
    float* o = out + (size_t)d * HIDDEN + c * 4;
    atomicAdd(o + 0, coef * v.x);
    atomicAdd(o + 1, coef * v.y);
    atomicAdd(o + 2, coef * v.z);
    atomicAdd(o + 3, coef * v.w);
}

// ---------------- GEMM2 via V_WMMA_F32_16X16X4_F32, BN1+ReLU fused on A load ----------------
// block = 256 threads (8 waves); block handles 16 nodes x 128 cols; wave w -> cols 16w..16w+15
__global__ __launch_bounds__(256) void gemm2_wmma_kernel(
        const float* __restrict__ hin,   // pre-BN layer-1 output [n,128]
        const float* __restrict__ w2,    // [128,128]
        const float* __restrict__ g, const float* __restrict__ be,
        const float* __restrict__ mean, const float* __restrict__ var,
        float* __restrict__ xw2, int n) {
    __shared__ float lA[16 * 132];   // padded stride vs 64 banks
    __shared__ float sc[HIDDEN];
    __shared__ float sh[HIDDEN];
    int t = threadIdx.x;
    if (t < HIDDEN) {
        float s = g[t] * rsqrtf(var[t] + BN_EPS);
        sc[t] = s;
        sh[t] = fmaf(-mean[t], s, be[t]);
    }
    __syncthreads();
    int row0 = blockIdx.x * 16;
    for (int i = t; i < 16 * HIDDEN; i += 256) {
        int r = i >> 7, col = i & 127;
        int node = row0 + r;
        float v = 0.0f;
        if (node < n)
            v = fmaxf(fmaf(sc[col], hin[(size_t)node * HIDDEN + col], sh[col]), 0.0f);
        lA[r * 132 + col] = v;
    }
    __syncthreads();

    int lane = t & 31;
    int nb   = (t >> 5) * 16;          // wave's output column base
    int m    = lane & 15;              // A row for this lane
    int kk   = (lane >> 4) * 2;        // lanes 16-31 hold k+2,k+3
    int ncol = nb + (lane & 15);       // B/D column for this lane

    v8f c = {};
#pragma unroll 8
    for (int k0 = 0; k0 < HIDDEN; k0 += 4) {
        v2f a, b;
        a.x = lA[m * 132 + k0 + kk];
        a.y = lA[m * 132 + k0 + kk + 1];
        b.x = w2[(k0 + kk) * HIDDEN + ncol];
        b.y = w2[(k0 + kk + 1) * HIDDEN + ncol];
        c = __builtin_amdgcn_wmma_f32_16x16x4_f32(false, a, false, b,
                                                  (short)0, c, false, false);
    }
    int mrow = (lane >> 4) * 8;        // C/D: VGPR v -> row v + 8*(lane>=16)
#pragma unroll
    for (int v = 0; v < 8; ++v) {
        int node = row0 + mrow + v;
        if (node < n) xw2[(size_t)node * HIDDEN + ncol] = c[v];
    }
}

// ---------------- BN2+ReLU + segment max pool (uint atomicMax, vals >= 0) ----------------
__global__ void pool_kernel(const float* __restrict__ hin, const long long* __restrict__ batch,
                            const float* __restrict__ g, const float* __restrict__ be,
                            const float* __restrict__ mean, const float* __restrict__ var,
                            float* __restrict__ pooled, int n) {
    int i = blockIdx.x * blockDim.x + threadIdx.x;
    if (i >= n * HIDDEN) return;
    int node = i >> 7, h = i & 127;
    float s = g[h] * rsqrtf(var[h] + BN_EPS);
    float v = fmaxf(fmaf(s, hin[i], fmaf(-mean[h], s, be[h])), 0.0f);
    int gi = (int)batch[node];
    atomicMax((unsigned int*)&pooled[(size_t)gi * HIDDEN + h], __float_as_uint(v));
}

// ---------------- final: sigmoid(pooled @ lin_w + lin_b) ----------------
__global__ __launch_bounds__(128) void final_kernel(const float* __restrict__ pooled,
                                                    const float* __restrict__ lw,
                                                    const float* __restrict__ lb,
                                                    float* __restrict__ out) {
    __shared__ float red[128];
    int gidx = blockIdx.x, t = threadIdx.x;
    red[t] = pooled[(size_t)gidx * HIDDEN + t] * lw[t];
    __syncthreads();
    for (int s = 64; s > 0; s >>= 1) {
        if (t < s) red[t] += red[t + s];
        __syncthreads();
    }
    if (t == 0) out[gidx] = 1.0f / (1.0f + expf(-(red[0] + lb[0])));
}

extern "C" void kernel_launch(void* const* d_in, const int* in_sizes, int n_in,
                              void* d_out, int out_size, void* d_ws, size_t ws_size,
                              hipStream_t stream) {
    const float*     x     = (const float*)d_in[0];
    const long long* edge  = (const long long*)d_in[1];
    const long long* batch = (const long long*)d_in[2];
    const float* w1  = (const float*)d_in[3];
    const float* b1  = (const float*)d_in[4];
    const float* g1  = (const float*)d_in[5];
    const float* be1 = (const float*)d_in[6];
    const float* m1  = (const float*)d_in[7];
    const float* v1  = (const float*)d_in[8];
    const float* w2  = (const float*)d_in[9];
    const float* b2  = (const float*)d_in[10];
    const float* g2  = (const float*)d_in[11];
    const float* be2 = (const float*)d_in[12];
    const float* m2  = (const float*)d_in[13];
    const float* v2  = (const float*)d_in[14];
    const float* lw  = (const float*)d_in[15];
    const float* lb  = (const float*)d_in[16];
    float* out = (float*)d_out;

    int n = in_sizes[0] / IN_DIM;   // 100000
    int e = in_sizes[1] / 2;        // 3200000
    const long long* src = edge;
    const long long* dst = edge + e;

    size_t per = (size_t)n * HIDDEN;
    float* bufA   = (float*)d_ws;                  // xw1, later xw2
    float* bufB   = bufA + per;                    // scat1, later scat2
    float* dis    = bufB + per;                    // deg -> rsqrt(deg)
    float* pooled = dis + (((size_t)n + 63) & ~(size_t)63);

    int pg = N_GRAPHS * HIDDEN;
    int initN = (n > pg) ? n : pg;
    size_t nh = (size_t)n * HIDDEN;
    size_t ew = (size_t)e * 32;

    init_kernel<<<(initN + 255) / 256, 256, 0, stream>>>(dis, pooled, n, pg);
    deg_kernel<<<(e + 255) / 256, 256, 0, stream>>>(dst, dis, e);
    dis_kernel<<<(n + 255) / 256, 256, 0, stream>>>(dis, n);

    // layer 1
    gemm1_kernel<<<(n + 7) / 8, 128, 0, stream>>>(x, w1, bufA, n);
    scat_init_kernel<<<(unsigned)((nh + 255) / 256), 256, 0, stream>>>(bufB, bufA, dis, b1, n);
    scat_edge_kernel<<<(unsigned)((ew + 255) / 256), 256, 0, stream>>>(bufB, bufA, dis, src, dst, e);

    // layer 2 (BN1+ReLU fused into GEMM2 A-tile load)
    gemm2_wmma_kernel<<<(n + 15) / 16, 256, 0, stream>>>(bufB, w2, g1, be1, m1, v1, bufA, n);
    scat_init_kernel<<<(unsigned)((nh + 255) / 256), 256, 0, stream>>>(bufB, bufA, dis, b2, n);
    scat_edge_kernel<<<(unsigned)((ew + 255) / 256), 256, 0, stream>>>(bufB, bufA, dis, src, dst, e);

    // BN2+ReLU fused pool, then head
    pool_kernel<<<(unsigned)((nh + 255) / 256), 256, 0, stream>>>(bufB, batch, g2, be2, m2, v2, pooled, n);
    final_kernel<<<N_GRAPHS, 128, 0, stream>>>(pooled, lw, lb, out);
}